// VolumeRenderer_43087111914123
// MI455X (gfx1250) — compile-verified
//
#include <hip/hip_runtime.h>

// NeRF volume-rendering integration (fancy_integration, clamp='relu').
// B=4, R=32768, S=64 -> 131072 rays x 64 samples.
//
// Roofline: 203MB total traffic @ 23.3TB/s => ~8.7us; ~1.3e8 FLOP => ~0.64
// FLOP/byte, purely HBM-bound. Strategy: touch every byte once with wide
// coalesced accesses, overlap the rgb stream (100MB, consumed last) with the
// per-ray scan using CDNA5 async global->LDS copies (ASYNCcnt path), and mark
// the whole single-pass stream non-temporal (TH=NT) so it does not churn the
// 192MB L2.
//
// Mapping: one wave32 per ray; lane l owns samples 2l and 2l+1.
//   1) issue async rgb (768B/ray) -> LDS  (b128 + b64, all 32 lanes, NT)
//   2) coalesced NT b64 loads of sigma/z, alpha = 1-exp(-delta*relu(sigma)),
//      factor f = (1-alpha)+eps = exp(-delta*relu(sigma))+eps
//   3) Hillis-Steele multiplicative scan across lanes -> exclusive cumprod
//   4) coalesced NT b64 store of weights
//   5) s_wait_asynccnt 0; weighted rgb/z sums from LDS; butterfly reduce;
//      lane 0 writes rgb_final(3) + depth(1).

#define WAVESZ 32

typedef float v2f __attribute__((ext_vector_type(2)));

static constexpr int kS             = 64;
static constexpr int kRgbF          = kS * 3;   // 192 floats = 768 B per ray
static constexpr int kWavesPerBlock = 8;

__global__ __launch_bounds__(kWavesPerBlock * WAVESZ)
void volume_render_kernel(const float* __restrict__ rgb,
                          const float* __restrict__ sigma,
                          const float* __restrict__ z_vals,
                          float* __restrict__ out_rgb,
                          float* __restrict__ out_depth,
                          float* __restrict__ out_w,
                          int n_rays) {
  __shared__ float smem[kWavesPerBlock * kRgbF];

  const int lane = threadIdx.x & (WAVESZ - 1);
  const int wave = threadIdx.x >> 5;
  const int ray  = blockIdx.x * kWavesPerBlock + wave;
  if (ray >= n_rays) return;   // wave-uniform; no block barriers used

  // ---- 1) async-stage this ray's 768B rgb block into LDS (non-temporal) ---
  float* lds_ray = &smem[wave * kRgbF];
  {
    const float* g0 = rgb + (size_t)ray * kRgbF + lane * 4;        // 32*16B
    unsigned l0 = (unsigned)(uintptr_t)lds_ray + (unsigned)(lane * 16);
    asm volatile("global_load_async_to_lds_b128 %0, %1, off th:TH_LOAD_NT"
                 :: "v"(l0), "v"(g0) : "memory");
    const float* g1 = rgb + (size_t)ray * kRgbF + 128 + lane * 2;  // 32*8B
    unsigned l1 = (unsigned)(uintptr_t)lds_ray + 512u + (unsigned)(lane * 8);
    asm volatile("global_load_async_to_lds_b64 %0, %1, off th:TH_LOAD_NT"
                 :: "v"(l1), "v"(g1) : "memory");
  }

  // ---- 2) alphas (NT streaming loads) -------------------------------------
  const size_t base = (size_t)ray * kS;
  const v2f sg = __builtin_nontemporal_load(
      reinterpret_cast<const v2f*>(sigma + base + 2 * lane));
  const v2f zz = __builtin_nontemporal_load(
      reinterpret_cast<const v2f*>(z_vals + base + 2 * lane));

  const float z_next = __shfl_down(zz.x, 1, WAVESZ);               // z[2l+2]
  const float d0 = zz.y - zz.x;
  const float d1 = (lane == WAVESZ - 1) ? 1e10f : (z_next - zz.y);

  const float t0 = __expf(-d0 * fmaxf(sg.x, 0.0f));  // = 1 - alpha0
  const float t1 = __expf(-d1 * fmaxf(sg.y, 0.0f));  // = 1 - alpha1
  const float a0 = 1.0f - t0, a1 = 1.0f - t1;
  const float f0 = t0 + 1e-10f, f1 = t1 + 1e-10f;    // (1 - alpha + eps)

  // ---- 3) exclusive cumprod via lane scan ---------------------------------
  float p = f0 * f1;                                 // local pair product
  #pragma unroll
  for (int off = 1; off < WAVESZ; off <<= 1) {
    const float v = __shfl_up(p, (unsigned)off, WAVESZ);
    if (lane >= off) p *= v;
  }
  float excl = __shfl_up(p, 1u, WAVESZ);             // exclusive over pairs
  if (lane == 0) excl = 1.0f;

  const float w0 = a0 * excl;                        // T(2l)   = excl
  const float w1 = a1 * (excl * f0);                 // T(2l+1) = excl*f0

  // ---- 4) weights out (coalesced NT b64 store) ----------------------------
  v2f wout; wout.x = w0; wout.y = w1;
  __builtin_nontemporal_store(wout,
      reinterpret_cast<v2f*>(out_w + base + 2 * lane));

  // ---- 5) weighted sums from LDS ------------------------------------------
  asm volatile("s_wait_asynccnt 0" ::: "memory");
  const float* L = lds_ray + 6 * lane;               // two samples x 3ch
  float ar = w0 * L[0] + w1 * L[3];
  float ag = w0 * L[1] + w1 * L[4];
  float ab = w0 * L[2] + w1 * L[5];
  float az = w0 * zz.x + w1 * zz.y;
  #pragma unroll
  for (int m = WAVESZ / 2; m >= 1; m >>= 1) {
    ar += __shfl_xor(ar, m, WAVESZ);
    ag += __shfl_xor(ag, m, WAVESZ);
    ab += __shfl_xor(ab, m, WAVESZ);
    az += __shfl_xor(az, m, WAVESZ);
  }
  if (lane == 0) {
    out_rgb[(size_t)ray * 3 + 0] = ar;
    out_rgb[(size_t)ray * 3 + 1] = ag;
    out_rgb[(size_t)ray * 3 + 2] = ab;
    out_depth[ray] = az;
  }
}

extern "C" void kernel_launch(void* const* d_in, const int* in_sizes, int n_in,
                              void* d_out, int out_size, void* d_ws, size_t ws_size,
                              hipStream_t stream) {
  (void)n_in; (void)out_size; (void)d_ws; (void)ws_size;
  const float* rgb   = (const float*)d_in[0];
  const float* sigma = (const float*)d_in[1];
  const float* z     = (const float*)d_in[2];
  const int n_rays   = in_sizes[1] / kS;                  // B*R (sigma is B*R*S)
  float* out_rgb     = (float*)d_out;                     // [B*R, 3]
  float* out_depth   = out_rgb + (size_t)n_rays * 3;      // [B*R, 1]
  float* out_w       = out_depth + n_rays;                // [B*R, S]
  const int blocks   = (n_rays + kWavesPerBlock - 1) / kWavesPerBlock;
  volume_render_kernel<<<blocks, kWavesPerBlock * WAVESZ, 0, stream>>>(
      rgb, sigma, z, out_rgb, out_depth, out_w, n_rays);
}